// WaveletLoss_34746285424740
// MI455X (gfx1250) — compile-verified
//
#include <hip/hip_runtime.h>

typedef float v2f __attribute__((ext_vector_type(2)));
typedef float v8f __attribute__((ext_vector_type(8)));

#define INV_SQRT2  0.70710678118654752f
#define INV_2SQRT2 0.35355339059327376f
#define THR1 0.19607843137254902f   // 50/255
#define THR2 0.09803921568627451f   // 25/255
#define THR3 0.04901960784313725f   // 12.5/255

// ---- composite 1D Haar operators on 16 points --------------------------------
// Plow rows: [P1(8); P2(4); P3(2); zeros(2)]   (lowpass composites per level)
// Qhigh rows:[Q1(8); Q2(4); Q3(2); zeros(2)]   (level-l highpass composites)
static __device__ __forceinline__ float haar_pl(int r, int c) {
  if (r < 8)  return (c == 2*r || c == 2*r + 1) ? INV_SQRT2 : 0.0f;
  if (r < 12) { int i = r - 8;  return (c >= 4*i && c < 4*i + 4) ? 0.5f : 0.0f; }
  if (r < 14) { int i = r - 12; return (c >= 8*i && c < 8*i + 8) ? INV_2SQRT2 : 0.0f; }
  return 0.0f;
}
static __device__ __forceinline__ float haar_qh(int r, int c) {
  if (r < 8) {
    if (c == 2*r)     return  INV_SQRT2;
    if (c == 2*r + 1) return -INV_SQRT2;
    return 0.0f;
  }
  if (r < 12) {
    int i = r - 8;
    if (c >= 4*i     && c < 4*i + 2) return  0.5f;
    if (c >= 4*i + 2 && c < 4*i + 4) return -0.5f;
    return 0.0f;
  }
  if (r < 14) {
    int i = r - 12;
    if (c >= 8*i     && c < 8*i + 4) return  INV_2SQRT2;
    if (c >= 8*i + 4 && c < 8*i + 8) return -INV_2SQRT2;
    return 0.0f;
  }
  return 0.0f;
}

static __device__ __forceinline__ float softt(float x, float t) {
  float m = fmaxf(fabsf(x) - t, 0.0f);
  return copysignf(m, x);
}

// value from lane ^ mask (wave32)
static __device__ __forceinline__ float lane_xor_f(float v, int mask) {
  int lane = (int)(threadIdx.x & 31u);
  int idx  = ((lane ^ mask) & 31) << 2;
  return __int_as_float(__builtin_amdgcn_ds_bpermute(idx, __float_as_int(v)));
}

// D = A(16x4) * B(4x16) + C, f32 WMMA
static __device__ __forceinline__ v8f wmma4(v2f a, v2f b, v8f c) {
  return __builtin_amdgcn_wmma_f32_16x16x4_f32(false, a, false, b, (short)0, c,
                                               false, false);
}

// Convert a C/D-layout 16x16 f32 result into four B-layout K=4 chunks.
// C layout: vgpr r holds row r (lanes 0-15) and row r+8 (lanes 16-31).
// B chunk k, vgpr j needs row 4k+2*hi+j at column lane&15.
struct Bmat { v2f ch[4]; };
static __device__ __forceinline__ Bmat c_to_b(const v8f C, bool hi) {
  float sw[8];
#pragma unroll
  for (int r = 0; r < 8; ++r) sw[r] = lane_xor_f(C[r], 16);
  Bmat b;
  v2f t;
  t.x = hi ? sw[2] : C[0];  t.y = hi ? sw[3] : C[1];  b.ch[0] = t;
  t.x = hi ? sw[6] : C[4];  t.y = hi ? sw[7] : C[5];  b.ch[1] = t;
  t.x = hi ? C[2]  : sw[0]; t.y = hi ? C[3]  : sw[1]; b.ch[2] = t;
  t.x = hi ? C[6]  : sw[4]; t.y = hi ? C[7]  : sw[5]; b.ch[3] = t;
  return b;
}

static __device__ __forceinline__ void accum(const v8f Yp, const v8f Yt,
                                             const int* lvl, const float* thr,
                                             float& a1, float& a2, float& a3) {
#pragma unroll
  for (int r = 0; r < 8; ++r) {
    float tr = thr[r];
    float d = fabsf(softt(Yp[r], tr) - softt(Yt[r], tr));
    a1 += (lvl[r] == 1) ? d : 0.0f;
    a2 += (lvl[r] == 2) ? d : 0.0f;
    a3 += (lvl[r] == 3) ? d : 0.0f;
  }
}

// 64 images of 512x512; 16x16 patches -> 32x32 patches/image, 65536 total.
// 1024 blocks * 8 waves = 8192 waves; 8 patches per wave.
__global__ __launch_bounds__(256) void wavelet_main(const float* __restrict__ pred,
                                                    const float* __restrict__ target,
                                                    float* __restrict__ partials) {
  const int lane = (int)(threadIdx.x & 31u);
  const int wv   = (int)(threadIdx.x >> 5);
  const int hi   = lane >> 4;
  const int r16  = lane & 15;

  // Constant operand chunks: pl/qh serve both as A-layout of Plow/Qhigh and
  // as B-layout of Plow^T/Qhigh^T (identical per-lane values).
  v2f pl[4], qh[4];
#pragma unroll
  for (int k = 0; k < 4; ++k) {
    int c0 = 4*k + 2*hi;
    v2f t;
    t.x = haar_pl(r16, c0); t.y = haar_pl(r16, c0 + 1); pl[k] = t;
    t.x = haar_qh(r16, c0); t.y = haar_qh(r16, c0 + 1); qh[k] = t;
  }

  // Per-element level id / threshold for C-layout results (diag blocks valid):
  // rows 0-7 x cols 0-7 -> level 1; 8-11 x 8-11 -> level 2; 12-13 x 12-13 -> 3.
  int lvl[8];
  float thr[8];
#pragma unroll
  for (int r = 0; r < 8; ++r) {
    int row = r + 8*hi, col = r16, l = 0;
    if (row < 8 && col < 8) l = 1;
    else if (row >= 8 && row < 12 && col >= 8 && col < 12) l = 2;
    else if (row >= 12 && row < 14 && col >= 12 && col < 14) l = 3;
    lvl[r] = l;
    thr[r] = (l == 1) ? THR1 : ((l == 2) ? THR2 : THR3);
  }

  float a1 = 0.0f, a2 = 0.0f, a3 = 0.0f;

  const int waveId = (int)blockIdx.x * ((int)blockDim.x >> 5) + wv;
#pragma unroll 1
  for (int i = 0; i < 8; ++i) {
    int pid = waveId * 8 + i;
    int b   = pid >> 10;          // 1024 patches per image
    int rem = pid & 1023;
    int py  = rem >> 5;
    int px  = rem & 31;
    size_t base = ((size_t)b << 18) + (size_t)(py*16 + r16) * 512 + (size_t)(px*16);
    const float* pp = pred + base;
    const float* pt = target + base;

    // X in A layout: lane holds row r16, cols 4k+2*hi .. +1
    v2f xap[4], xat[4];
#pragma unroll
    for (int k = 0; k < 4; ++k) {
      int c0 = 4*k + 2*hi;
      xap[k] = *reinterpret_cast<const v2f*>(pp + c0);
      xat[k] = *reinterpret_cast<const v2f*>(pt + c0);
    }

    // R = X * Plow^T, S = X * Qhigh^T for both images
    v8f Rp = {}, Sp = {}, Rt = {}, St = {};
#pragma unroll
    for (int k = 0; k < 4; ++k) {
      Rp = wmma4(xap[k], pl[k], Rp);
      Sp = wmma4(xap[k], qh[k], Sp);
      Rt = wmma4(xat[k], pl[k], Rt);
      St = wmma4(xat[k], qh[k], St);
    }

    Bmat bRp = c_to_b(Rp, hi != 0), bSp = c_to_b(Sp, hi != 0);
    Bmat bRt = c_to_b(Rt, hi != 0), bSt = c_to_b(St, hi != 0);

    // Y_A = Plow * S  (cV blocks)
    v8f Yp = {}, Yt = {};
#pragma unroll
    for (int k = 0; k < 4; ++k) { Yp = wmma4(pl[k], bSp.ch[k], Yp);
                                  Yt = wmma4(pl[k], bSt.ch[k], Yt); }
    accum(Yp, Yt, lvl, thr, a1, a2, a3);

    // Y_B = Qhigh * R  (cH blocks)
    Yp = (v8f){}; Yt = (v8f){};
#pragma unroll
    for (int k = 0; k < 4; ++k) { Yp = wmma4(qh[k], bRp.ch[k], Yp);
                                  Yt = wmma4(qh[k], bRt.ch[k], Yt); }
    accum(Yp, Yt, lvl, thr, a1, a2, a3);

    // Y_C = Qhigh * S  (cD blocks)
    Yp = (v8f){}; Yt = (v8f){};
#pragma unroll
    for (int k = 0; k < 4; ++k) { Yp = wmma4(qh[k], bSp.ch[k], Yp);
                                  Yt = wmma4(qh[k], bSt.ch[k], Yt); }
    accum(Yp, Yt, lvl, thr, a1, a2, a3);
  }

  // wave32 butterfly reduction
#pragma unroll
  for (int off = 16; off >= 1; off >>= 1) {
    a1 += lane_xor_f(a1, off);
    a2 += lane_xor_f(a2, off);
    a3 += lane_xor_f(a3, off);
  }

  __shared__ float red[8][3];
  if (lane == 0) { red[wv][0] = a1; red[wv][1] = a2; red[wv][2] = a3; }
  __syncthreads();
  if (threadIdx.x < 3) {
    float s = 0.0f;
#pragma unroll
    for (int w = 0; w < 8; ++w) s += red[w][threadIdx.x];
    partials[blockIdx.x * 3 + threadIdx.x] = s;
  }
}

__global__ __launch_bounds__(256) void wavelet_finalize(const float* __restrict__ partials,
                                                        float* __restrict__ out,
                                                        int nblocks) {
  __shared__ float sm[256][3];
  float s0 = 0.0f, s1 = 0.0f, s2 = 0.0f;
  for (int i = (int)threadIdx.x; i < nblocks; i += (int)blockDim.x) {
    s0 += partials[i*3 + 0];
    s1 += partials[i*3 + 1];
    s2 += partials[i*3 + 2];
  }
  sm[threadIdx.x][0] = s0; sm[threadIdx.x][1] = s1; sm[threadIdx.x][2] = s2;
  __syncthreads();
  if (threadIdx.x == 0) {
    float t0 = 0.0f, t1 = 0.0f, t2 = 0.0f;
    for (int j = 0; j < 256; ++j) { t0 += sm[j][0]; t1 += sm[j][1]; t2 += sm[j][2]; }
    const float N1 = 64.0f * 256.0f * 256.0f;   // coeffs per subband, level 1
    const float N2 = 64.0f * 128.0f * 128.0f;
    const float N3 = 64.0f * 64.0f  * 64.0f;
    out[0] = t0 / (3.0f * N1)
           + 0.5f * t1 / (3.0f * N2)
           + (1.0f / 3.0f) * t2 / (3.0f * N3);
  }
}

extern "C" void kernel_launch(void* const* d_in, const int* in_sizes, int n_in,
                              void* d_out, int out_size, void* d_ws, size_t ws_size,
                              hipStream_t stream) {
  (void)in_sizes; (void)n_in; (void)out_size; (void)ws_size;
  const float* pred   = (const float*)d_in[0];
  const float* target = (const float*)d_in[1];
  float* out      = (float*)d_out;
  float* partials = (float*)d_ws;   // 1024 * 3 floats = 12 KB

  wavelet_main<<<dim3(1024), dim3(256), 0, stream>>>(pred, target, partials);
  wavelet_finalize<<<dim3(1), dim3(256), 0, stream>>>(partials, out, 1024);
}